// MyModel_1099511627938
// MI455X (gfx1250) — compile-verified
//
#include <hip/hip_runtime.h>

// v_wmma_f32_16x16x4_f32 fragment types (wave32):
//   A: 16x4 f32  -> 2 VGPRs/lane
//   B: 4x16 f32  -> 2 VGPRs/lane
//   C/D: 16x16 f32 -> 8 VGPRs/lane
typedef __attribute__((ext_vector_type(2))) float v2f;
typedef __attribute__((ext_vector_type(8))) float v8f;

#define NBLOCK            256
#define NTHREADS          256
#define WAVES_PER_BLOCK   (NTHREADS / 32)
#define NWAVES            (NBLOCK * WAVES_PER_BLOCK)

// Each wave accumulates a 16x16 f32 tile (only the top-left 5x5 is live) over
// a contiguous slice of the K dimension using V_WMMA_F32_16X16X4_F32.
//
// Correctness notes:
//  - Dead fragment lanes (m>=5 rows of A, n>=5 cols of B) are address-clamped,
//    NOT value-zeroed. Every C element is an independent dot product, so
//    garbage in dead lanes only lands in C entries outside the 5x5 live
//    region, which are never read.
//  - Assumes K even (K = 4e6 here) so x rows are float2-aligned; any K%4 tail
//    is folded in scalar form by reduce_partials.
__global__ __launch_bounds__(NTHREADS)
void skinny_gemm_wmma(const float* __restrict__ x,   // [5, K] row-major
                      const float* __restrict__ y,   // [K, 5] row-major
                      float* __restrict__ partial,   // [NBLOCK, 25]
                      int K)                          // fits int32
{
    const int lane        = threadIdx.x & 31;
    const int wave_in_blk = threadIdx.x >> 5;

    // Wave index is uniform within the wave; readfirstlane keeps the loop
    // bounds in SGPRs (scalar branch loop, no exec-mask loop).
    const int w = __builtin_amdgcn_readfirstlane(
                      (int)blockIdx.x * WAVES_PER_BLOCK + wave_in_blk);

    const int kIters = K >> 2;                       // WMMA steps (k/4)
    // Round per-wave work to a multiple of the unroll factor so the scalar
    // tail loop (unroll remainder) almost never runs.
    const int perWave = (((kIters + NWAVES - 1) / NWAVES) + 3) & ~3;
    int it    = w * perWave;
    int itEnd = it + perWave;
    if (itEnd > kIters) itEnd = kIters;

    // ISA fragment layout (32-bit A 16x4 / B 4x16):
    //   lanes 0-15 : VGPR0=K0, VGPR1=K1 ; lanes 16-31 : VGPR0=K2, VGPR1=K3
    const int  half = lane >> 4;        // which K pair this half-wave holds
    const int  sub  = lane & 15;        // m for A, n for B
    const bool live = (sub < 5);
    const int  srow = live ? sub : 0;   // address clamp keeps dead lanes in-bounds

    v8f c = {};

    if (it < itEnd) {
        // 32-bit element indices (arrays are 20M floats = 80 MB): uniform SGPR
        // base + 32-bit VGPR offset with scale_offset addressing, so the only
        // per-iteration address work is two v_add_nc_u32.
        const float2* __restrict__ x2 = (const float2*)x;   // [5, K/2]
        unsigned xoff2 = (unsigned)srow * ((unsigned)K >> 1)    // row base (float2s)
                       + ((unsigned)it << 1) + (unsigned)half;  // k pair index
        unsigned yoff  = (((unsigned)it << 2) + 2u * half) * 5u + (unsigned)srow;

        #pragma unroll 4
        for (; it < itEnd; ++it) {
            const float2 xa  = x2[xoff2];      // A: k,k+1 contiguous -> one b64
            const float  yb0 = y[yoff];        // B: y[k][n]   (coalesced 64B/half-wave)
            const float  yb1 = y[yoff + 5];    // B: y[k+1][n] (coalesced 64B/half-wave)

            v2f a, b;
            a[0] = xa.x;  a[1] = xa.y;
            b[0] = yb0;   b[1] = yb1;

            // D = A*B + C  (EXEC is all ones here)
            c = __builtin_amdgcn_wmma_f32_16x16x4_f32(
                    /*neg_a=*/false, a, /*neg_b=*/false, b,
                    /*c_mod=*/(short)0, c, /*reuse_a=*/false, /*reuse_b=*/false);

            xoff2 += 2u;    // 4 k = 2 float2s
            yoff  += 20u;   // 4 k * 5 cols
        }
    }

    // ---- Deterministic block reduction over the 8 waves ----
    // C layout: element (m, n) with m < 8 lives in VGPR m at lane n (lanes 0-15).
    __shared__ float red[WAVES_PER_BLOCK * 25];
    if (half == 0 && live) {            // lane == n in [0,5)
        #pragma unroll
        for (int m = 0; m < 5; ++m)
            red[wave_in_blk * 25 + m * 5 + sub] = c[m];
    }
    __syncthreads();

    if (threadIdx.x < 25) {
        float s = 0.0f;
        #pragma unroll
        for (int wv = 0; wv < WAVES_PER_BLOCK; ++wv)
            s += red[wv * 25 + threadIdx.x];
        partial[blockIdx.x * 25 + threadIdx.x] = s;
    }
}

// Fold the 256 per-block partials in a fixed order (deterministic), apply the
// reference's x2, and handle any K % 4 scalar tail.
__global__ void reduce_partials(const float* __restrict__ partial,
                                const float* __restrict__ x,
                                const float* __restrict__ y,
                                float* __restrict__ out,
                                int K)
{
    const int t = threadIdx.x;
    if (t < 25) {
        float s = 0.0f;
        for (int b = 0; b < NBLOCK; ++b)
            s += partial[b * 25 + t];

        // scalar tail for K not divisible by 4 (none for K = 4M, kept for safety)
        const int m = t / 5, n = t % 5;
        for (int k = (K >> 2) << 2; k < K; ++k)
            s += x[(long long)m * K + k] * y[(long long)k * 5 + n];

        out[t] = 2.0f * s;
    }
}

extern "C" void kernel_launch(void* const* d_in, const int* in_sizes, int n_in,
                              void* d_out, int out_size, void* d_ws, size_t ws_size,
                              hipStream_t stream)
{
    // inputs: d_in[0] = graph (unused int), d_in[1] = x [5,K] f32, d_in[2] = y [K,5] f32
    const float* x = (const float*)d_in[1];
    const float* y = (const float*)d_in[2];
    float* out     = (float*)d_out;          // 25 floats
    float* partial = (float*)d_ws;           // NBLOCK * 25 floats = 25.6 KB scratch

    const int K = in_sizes[1] / 5;

    skinny_gemm_wmma<<<NBLOCK, NTHREADS, 0, stream>>>(x, y, partial, K);
    reduce_partials<<<1, 32, 0, stream>>>(partial, x, y, out, K);
}